// SPR_W_B_56977036148941
// MI455X (gfx1250) — compile-verified
//
#include <hip/hip_runtime.h>
#include <math.h>

// ---------------------------------------------------------------------------
// SPR (phase retrieval) pipeline for MI455X (gfx1250, wave32).
//   - batched 512x512 fft2/ifft2 via four-step 32x16 decomposition:
//       stage1: 16 parallel Stockham radix-2 FFT-32 in LDS
//       stage2: 512-point twiddle
//       stage3: DFT-16 as 16x16 matmul via V_WMMA_F32_16X16X4_F32 (fp32 WMMA)
//   - one wave per 512-point line, 8 waves (256 thr) per block, 64KB LDS
//   - top-4096 threshold: 4-pass MSB radix select on |B|^2 bit patterns
// ---------------------------------------------------------------------------

#define PI_F 3.14159265358979323846f

static constexpr int   NIMG  = 8;
static constexpr int   MDIM  = 512;
static constexpr int   TPIX  = MDIM * MDIM;        // 262144
static constexpr int   NPIX  = NIMG * MDIM * MDIM; // 2097152
static constexpr int   NITER = 6;
static constexpr int   BETA  = 4096;
static constexpr float LAM   = 0.1f;
static constexpr float WLINV = 1.0f / 6.37e-07f;

typedef __attribute__((ext_vector_type(2))) float v2f;
typedef __attribute__((ext_vector_type(8))) float v8f;

struct cpx { float x, y; };
__device__ __forceinline__ cpx cmul(cpx a, cpx b) {
  return cpx{a.x * b.x - a.y * b.y, a.x * b.y + a.y * b.x};
}
__device__ __forceinline__ cpx cadd(cpx a, cpx b) { return cpx{a.x + b.x, a.y + b.y}; }
__device__ __forceinline__ cpx csub(cpx a, cpx b) { return cpx{a.x - b.x, a.y - b.y}; }

// ---------------------------------------------------------------------------
// One 512-point FFT per wave.  dir = -1 forward, +1 inverse.
// rows: rowmul=512, stride=1 ; cols: rowmul=1, stride=512 (in-place safe).
// ---------------------------------------------------------------------------
__global__ __launch_bounds__(256)
void k_fft_pass(const cpx* __restrict__ src, cpx* __restrict__ dst,
                int rowmul, int stride, float dir, float scale)
{
  __shared__ cpx lds[2][8][MDIM];   // ping-pong, 64 KB

  const int wave = threadIdx.x >> 5;
  const int lane = threadIdx.x & 31;
  const int line = blockIdx.x * 8 + wave;      // 4096 lines total
  const int img  = line >> 9;
  const int r    = line & 511;
  const int base = img * TPIX + r * rowmul;

  cpx* cur = &lds[0][wave][0];
  cpx* nxt = &lds[1][wave][0];

  for (int i = lane; i < MDIM; i += 32)
    cur[i] = src[base + i * stride];
  __syncthreads();

  // ---- stage 1: 16 interleaved FFT-32 (Stockham radix-2, auto-sorting) ----
  // element (n1, n2) lives at LDS index n1 + 16*n2
  int Ns = 1;
  #pragma unroll
  for (int stage = 0; stage < 5; ++stage) {
    for (int t = lane; t < 256; t += 32) {
      int n1 = t & 15;
      int j  = t >> 4;                       // butterfly 0..15
      cpx v0 = cur[n1 + 16 * j];
      cpx v1 = cur[n1 + 16 * (j + 16)];
      float ang = dir * 2.0f * PI_F * (float)(j % Ns) / (float)(2 * Ns);
      float s, c; __sincosf(ang, &s, &c);
      cpx w1 = cmul(v1, cpx{c, s});
      int idxD = (j / Ns) * (2 * Ns) + (j % Ns);
      nxt[n1 + 16 * idxD]        = cadd(v0, w1);
      nxt[n1 + 16 * (idxD + Ns)] = csub(v0, w1);
    }
    cpx* tswap = cur; cur = nxt; nxt = tswap;
    Ns <<= 1;
    __syncthreads();
  }

  // ---- stage 2: twiddle  Y'[n1][k2] *= exp(i*dir*2pi*n1*k2/512) ----
  for (int i = lane; i < MDIM; i += 32) {
    int n1 = i & 15, k2 = i >> 4;
    float ang = dir * 2.0f * PI_F * (float)(n1 * k2) * (1.0f / 512.0f);
    float s, c; __sincosf(ang, &s, &c);
    cur[i] = cmul(cur[i], cpx{c, s});
  }
  __syncthreads();

  // ---- stage 3: X[(16b+N) + 32*M] = sum_K F16[M][K] * Y'[K][16b+N] --------
  // complex 16x16 x 16x32 matmul via V_WMMA_F32_16X16X4_F32 chains.
  const int half = (lane >= 16) ? 1 : 0;
  const int Mrow = lane & 15;
  const int col  = lane & 15;

  // A-matrix tiles (16x4 fp32 layout: VGPR0 = K+0 / K+2 per half-wave)
  v2f ar[4], ai[4], an[4];
  #pragma unroll
  for (int t = 0; t < 4; ++t) {
    int k0 = 4 * t + 2 * half;
    float th0 = dir * 2.0f * PI_F * (float)(Mrow * k0)       * (1.0f / 16.0f);
    float th1 = dir * 2.0f * PI_F * (float)(Mrow * (k0 + 1)) * (1.0f / 16.0f);
    float s0, c0, s1, c1;
    __sincosf(th0, &s0, &c0);
    __sincosf(th1, &s1, &c1);
    ar[t][0] = c0;  ar[t][1] = c1;   // Re(F)
    ai[t][0] = s0;  ai[t][1] = s1;   // Im(F)
    an[t][0] = -s0; an[t][1] = -s1;  // -Im(F)
  }

  #pragma unroll
  for (int b = 0; b < 2; ++b) {
    v8f accr = {0.f, 0.f, 0.f, 0.f, 0.f, 0.f, 0.f, 0.f};
    v8f acci = {0.f, 0.f, 0.f, 0.f, 0.f, 0.f, 0.f, 0.f};
    #pragma unroll
    for (int t = 0; t < 4; ++t) {
      int krow = 4 * t + 2 * half;
      int n    = col + 16 * b;
      cpx y0 = cur[(krow)     + 16 * n];
      cpx y1 = cur[(krow + 1) + 16 * n];
      v2f br = { y0.x, y1.x };
      v2f bi = { y0.y, y1.y };
      // D_re += Fr*Yr - Fi*Yi ; D_im += Fr*Yi + Fi*Yr
      accr = __builtin_amdgcn_wmma_f32_16x16x4_f32(false, ar[t], false, br, (short)0, accr, false, false);
      accr = __builtin_amdgcn_wmma_f32_16x16x4_f32(false, an[t], false, bi, (short)0, accr, false, false);
      acci = __builtin_amdgcn_wmma_f32_16x16x4_f32(false, ar[t], false, bi, (short)0, acci, false, false);
      acci = __builtin_amdgcn_wmma_f32_16x16x4_f32(false, ai[t], false, br, (short)0, acci, false, false);
    }
    // C/D layout: VGPR r holds M = r + 8*half, N = lane&15
    #pragma unroll
    for (int rr = 0; rr < 8; ++rr) {
      int k1 = rr + 8 * half;
      int k  = (16 * b + col) + 32 * k1;
      dst[base + k * stride] = cpx{accr[rr] * scale, acci[rr] * scale};
    }
  }
}

// ---------------------------------------------------------------------------
// Pointwise kernels
// ---------------------------------------------------------------------------
__global__ void k_T(const float* __restrict__ Q, const float* __restrict__ zp,
                    cpx* __restrict__ T)
{
  int i = blockIdx.x * blockDim.x + threadIdx.x;
  if (i >= TPIX) return;
  float a = zp[0] * WLINV;
  float s, c; __sincosf(a * Q[i], &s, &c);
  T[i] = cpx{c, s};
}

__global__ void k_init(cpx* __restrict__ B, cpx* __restrict__ X)
{
  int i = blockIdx.x * blockDim.x + threadIdx.x;
  if (i >= NPIX) return;
  // B0 = fft2(ones) = (M*M) * delta at [0,0] per image
  B[i] = ((i & (TPIX - 1)) == 0) ? cpx{(float)TPIX, 0.f} : cpx{0.f, 0.f};
  X[i] = cpx{0.f, 0.f};
}

__global__ void k_mulT(cpx* __restrict__ A, const cpx* __restrict__ T, float conjSign)
{
  int i = blockIdx.x * blockDim.x + threadIdx.x;
  if (i >= NPIX) return;
  cpx t = T[i & (TPIX - 1)];
  t.y *= conjSign;
  A[i] = cmul(A[i], t);
}

__global__ void k_W(cpx* __restrict__ W, const cpx* __restrict__ cXT,
                    const cpx* __restrict__ ifb)
{
  int i = blockIdx.x * blockDim.x + threadIdx.x;
  if (i >= NPIX) return;
  cpx v = cadd(cXT[i], ifb[i]);
  float m = sqrtf(v.x * v.x + v.y * v.y);
  float d = (m == 0.f) ? 1.f : m;
  W[i] = cpx{v.x / d, v.y / d};
}

__global__ void k_Qc(cpx* __restrict__ Qc, const cpx* __restrict__ H,
                     const cpx* __restrict__ W)
{
  int i = blockIdx.x * blockDim.x + threadIdx.x;
  if (i >= NPIX) return;
  Qc[i] = cmul(H[i], W[i]);
}

__global__ void k_sub(cpx* __restrict__ d, const cpx* __restrict__ a,
                      const cpx* __restrict__ b)
{
  int i = blockIdx.x * blockDim.x + threadIdx.x;
  if (i >= NPIX) return;
  d[i] = csub(a[i], b[i]);
}

// top-BETA threshold on |B|^2: 4-pass MSB-first radix select, single block.
__global__ __launch_bounds__(1024)
void k_topk(const cpx* __restrict__ B, float* __restrict__ thr2)
{
  __shared__ unsigned int hist[256];
  __shared__ unsigned int sPrefix, sK;
  if (threadIdx.x == 0) { sPrefix = 0u; sK = (unsigned)BETA; }
  __syncthreads();

  for (int p = 3; p >= 0; --p) {
    for (int h = threadIdx.x; h < 256; h += 1024) hist[h] = 0u;
    __syncthreads();
    unsigned int prefix = sPrefix;
    unsigned int maskHi = (p == 3) ? 0u : (0xFFFFFFFFu << (8 * (p + 1)));
    for (int i = threadIdx.x; i < NPIX; i += 1024) {
      cpx v = B[i];
      float m2 = v.x * v.x + v.y * v.y;          // >= 0 -> bit pattern monotone
      unsigned int u = __float_as_uint(m2);
      if ((u & maskHi) == (prefix & maskHi))
        atomicAdd(&hist[(u >> (8 * p)) & 255u], 1u);
    }
    __syncthreads();
    if (threadIdx.x == 0) {
      unsigned int k = sK, cum = 0u; int bsel = 0;
      for (int bb = 255; bb >= 0; --bb) {
        if (cum + hist[bb] >= k) { bsel = bb; break; }
        cum += hist[bb];
      }
      sPrefix = prefix | ((unsigned int)bsel << (8 * p));
      sK = k - cum;
    }
    __syncthreads();
  }
  if (threadIdx.x == 0) thr2[0] = __uint_as_float(sPrefix);
}

__global__ void k_mask(cpx* __restrict__ B, const float* __restrict__ thr2)
{
  int i = blockIdx.x * blockDim.x + threadIdx.x;
  if (i >= NPIX) return;
  cpx v = B[i];
  float m2 = v.x * v.x + v.y * v.y;
  if (!(m2 >= thr2[0])) B[i] = cpx{0.f, 0.f};
}

__global__ void k_X(cpx* __restrict__ X, const cpx* __restrict__ inp1, int applyLam)
{
  int i = blockIdx.x * blockDim.x + threadIdx.x;
  if (i >= NPIX) return;
  cpx v = inp1[i];
  if (applyLam) {
    float m2 = v.x * v.x + v.y * v.y;
    if (!(m2 > LAM * LAM)) v = cpx{0.f, 0.f};
  }
  X[i] = v;
}

// ---------------------------------------------------------------------------
extern "C" void kernel_launch(void* const* d_in, const int* in_sizes, int n_in,
                              void* d_out, int out_size, void* d_ws, size_t ws_size,
                              hipStream_t stream)
{
  (void)in_sizes; (void)n_in; (void)out_size; (void)ws_size;

  const cpx*   H  = (const cpx*)d_in[0];   // (8,512,512,2) f32 == interleaved complex
  const float* Q  = (const float*)d_in[1]; // (512,512)
  const float* zp = (const float*)d_in[2]; // scalar

  char* ws = (char*)d_ws;
  size_t off = 0;
  auto carve = [&](size_t nelem) -> cpx* {
    cpx* p = (cpx*)(ws + off);
    off += nelem * sizeof(cpx);
    return p;
  };
  cpx* T   = carve(TPIX);
  cpx* W   = carve(NPIX);
  cpx* B   = carve(NPIX);
  cpx* X   = carve(NPIX);
  cpx* Qc  = carve(NPIX);
  cpx* cXT = carve(NPIX);
  cpx* t1  = carve(NPIX);
  cpx* t2  = carve(NPIX);
  float* thr2 = (float*)(ws + off);

  const dim3 pb(256), pg((NPIX + 255) / 256);
  const dim3 fb(256), fg(4096 / 8);  // 4096 lines, 8 waves/block

  auto fft2 = [&](const cpx* s, cpx* d, float dir, float sc) {
    k_fft_pass<<<fg, fb, 0, stream>>>(s, d, /*rowmul*/MDIM, /*stride*/1, dir, sc); // rows
    k_fft_pass<<<fg, fb, 0, stream>>>(d, d, /*rowmul*/1, /*stride*/MDIM, dir, sc); // cols
  };
  const float ISC = 1.0f / 512.0f;  // per-pass inverse scale -> 1/(512*512) total

  k_T<<<dim3((TPIX + 255) / 256), pb, 0, stream>>>(Q, zp, T);
  k_init<<<pg, pb, 0, stream>>>(B, X);

  for (int it = 0; it < NITER; ++it) {
    // convXT = ifft2(fft2(X) * T)
    fft2(X, t1, -1.f, 1.f);
    k_mulT<<<pg, pb, 0, stream>>>(t1, T, +1.f);
    fft2(t1, cXT, +1.f, ISC);
    // W = normalize(convXT + ifft2(B))
    fft2(B, t2, +1.f, ISC);
    k_W<<<pg, pb, 0, stream>>>(W, cXT, t2);
    // Qc = H * W
    k_Qc<<<pg, pb, 0, stream>>>(Qc, H, W);
    // B = fft2(Qc - convXT), then top-BETA support selection
    k_sub<<<pg, pb, 0, stream>>>(t1, Qc, cXT);
    fft2(t1, B, -1.f, 1.f);
    k_topk<<<dim3(1), dim3(1024), 0, stream>>>(B, thr2);
    k_mask<<<pg, pb, 0, stream>>>(B, thr2);

    if (it < NITER - 1) {
      // inp1 = conv(Qc - ifft2(B), conj(T)); X = inp1 * (|inp1| > LAM)
      fft2(B, t2, +1.f, ISC);
      k_sub<<<pg, pb, 0, stream>>>(t1, Qc, t2);
      fft2(t1, t2, -1.f, 1.f);
      k_mulT<<<pg, pb, 0, stream>>>(t2, T, -1.f);     // * conj(T)
      fft2(t2, t1, +1.f, ISC);
      k_X<<<pg, pb, 0, stream>>>(X, t1, 1);
    }
  }

  hipMemcpyAsync(d_out, W, (size_t)NPIX * sizeof(cpx),
                 hipMemcpyDeviceToDevice, stream);
  hipMemcpyAsync((char*)d_out + (size_t)NPIX * sizeof(cpx), B,
                 (size_t)NPIX * sizeof(cpx), hipMemcpyDeviceToDevice, stream);
}